// Conv2d_86191403696259
// MI455X (gfx1250) — compile-verified
//
#include <hip/hip_runtime.h>

typedef __attribute__((ext_vector_type(2))) float v2f;
typedef __attribute__((ext_vector_type(8))) float v8f;

#define CW 2048
#define CH 2048
#define CHW (2048 * 2048)
#define KTOT 27
#define NJ 7 /* ceil(28/4) K-steps of the 16x16x4 f32 WMMA */

__global__ __launch_bounds__(256) void conv3x3_wmma_f32(
    const float* __restrict__ X,   // [3, 2048, 2048]
    const float* __restrict__ Wt,  // [16, 27]  (OC, IC*KH*KW)
    float* __restrict__ out)       // [16, 2048, 2048]
{
    const int lane = threadIdx.x & 31;
    const int waveInBlk = threadIdx.x >> 5;
    const int gw = blockIdx.x * (blockDim.x >> 5) + waveInBlk;
    const int nwaves = gridDim.x * (blockDim.x >> 5);

    const int m  = lane & 15;          // M row (A) / N column (B,C,D)
    const int kb = (lane >> 4) << 1;   // K sub-offset: 0 for lanes<16, 2 for lanes>=16

    // ---- Preload A (weights) once per wave; precompute B gather geometry ----
    v2f a[NJ];
    int colbase[NJ][2];  // ic*HW + (kh-1)*W + (kw-1)   (X_addr = colbase + p)
    int dy[NJ][2];
    int dx[NJ][2];
#pragma unroll
    for (int j = 0; j < NJ; ++j) {
        float av0 = 0.0f, av1 = 0.0f;
#pragma unroll
        for (int e = 0; e < 2; ++e) {
            const int k = 4 * j + kb + e;
            float w = 0.0f;
            if (k < KTOT) {
                w = Wt[m * KTOT + k];
                const int ic = k / 9;
                const int r  = k - 9 * ic;
                const int kh = r / 3;
                const int kw = r - 3 * kh;
                dy[j][e] = kh - 1;
                dx[j][e] = kw - 1;
                colbase[j][e] = ic * CHW + (kh - 1) * CW + (kw - 1);
            } else {
                // K padding slot: A = 0 kills any (finite) B value; point the
                // B load at a always-in-bounds location (dy=dx=0, colbase=0).
                dy[j][e] = 0;
                dx[j][e] = 0;
                colbase[j][e] = 0;
            }
            if (e == 0) av0 = w; else av1 = w;
        }
        a[j].x = av0;
        a[j].y = av1;
    }

    const int ntiles = CHW / 16;  // 262144 tiles of 16 pixels (along W)
    for (int t = gw; t < ntiles; t += nwaves) {
        const int p  = (t << 4) + m;     // this lane's output pixel index
        const int y  = p >> 11;          // row   (W == 2048)
        const int x  = p & (CW - 1);     // col
        const int h  = t >> 7;           // tile row (uniform in wave)
        const int wt = t & 127;          // tile col (uniform in wave)

        // ---- Gather B (im2col columns), fast path for interior tiles ----
        v2f bb[NJ];
        const bool interior = (h >= 1) & (h < CH - 1) & (wt >= 1) & (wt < 127);
        if (interior) {
#pragma unroll
            for (int j = 0; j < NJ; ++j) {
                bb[j].x = X[colbase[j][0] + p];
                bb[j].y = X[colbase[j][1] + p];
            }
        } else {
#pragma unroll
            for (int j = 0; j < NJ; ++j) {
#pragma unroll
                for (int e = 0; e < 2; ++e) {
                    const int yy = y + dy[j][e];
                    const int xx = x + dx[j][e];
                    const bool ok = ((unsigned)yy < (unsigned)CH) &
                                    ((unsigned)xx < (unsigned)CW);
                    const float v = ok ? X[colbase[j][e] + p] : 0.0f;
                    if (e == 0) bb[j].x = v; else bb[j].y = v;
                }
            }
        }

        // ---- 7 chained f32 WMMAs: D(16x16) += A(16x4) x B(4x16) ----
        v8f c = {};
#pragma unroll
        for (int j = 0; j < NJ; ++j) {
            c = __builtin_amdgcn_wmma_f32_16x16x4_f32(
                /*neg_a=*/false, a[j], /*neg_b=*/false, bb[j],
                /*c_mod=*/(short)0, c, /*reuse_a=*/false, /*reuse_b=*/false);
        }

        // ---- Store: VGPR v -> OC = v + 8*(lane>=16), pixel p ----
        const int ocbase = (lane >> 4) << 3;
#pragma unroll
        for (int v = 0; v < 8; ++v) {
            out[(ocbase + v) * CHW + p] = c[v];
        }
    }
}

extern "C" void kernel_launch(void* const* d_in, const int* in_sizes, int n_in,
                              void* d_out, int out_size, void* d_ws, size_t ws_size,
                              hipStream_t stream) {
    const float* x  = (const float*)d_in[0];   // 3*2048*2048 f32
    const float* wt = (const float*)d_in[1];   // 16*3*3*3    f32
    float* out = (float*)d_out;                // 16*2048*2048 f32

    dim3 grid(4096);   // 4096 blocks * 8 waves = 32768 waves -> 8 tiles/wave
    dim3 block(256);
    hipLaunchKernelGGL(conv3x3_wmma_f32, grid, block, 0, stream, x, wt, out);
}